// MaskGenerator_57148834840621
// MI455X (gfx1250) — compile-verified
//
#include <hip/hip_runtime.h>
#include <math.h>

// Problem constants (fixed by the reference)
#define BQ   2048
#define VQ   50000
#define DQ   512
#define TOPK 5

typedef __attribute__((ext_vector_type(2))) float v2f;
typedef __attribute__((ext_vector_type(8))) float v8f;

// ---------------------------------------------------------------------------
// Kernel 1: per-row top-5 of sp_z[b, :]  (streaming, bandwidth-bound)
// One block per row; 256 threads; float4 (global_load_b128) streaming.
// Each thread keeps a sorted-descending top-5 in registers, then an LDS tree
// merge produces the row's global top-5 (stable: value desc, index asc).
// ---------------------------------------------------------------------------
__global__ __launch_bounds__(256) void topk5_kernel(const float* __restrict__ sp_z,
                                                    int* __restrict__ idx_out) {
  const int b   = blockIdx.x;
  const int tid = threadIdx.x;
  const float4* row = reinterpret_cast<const float4*>(sp_z + (size_t)b * VQ);

  float tv[TOPK];
  int   ti[TOPK];
#pragma unroll
  for (int q = 0; q < TOPK; ++q) { tv[q] = -INFINITY; ti[q] = 0x7fffffff; }

  const int n4 = VQ / 4;  // 12500 float4s per row
#pragma unroll 4
  for (int j = tid; j < n4; j += 256) {
    if (j + 512 < n4) __builtin_prefetch(row + j + 512, 0, 0);  // global_prefetch_b8
    float4 v = row[j];
    const int base = j * 4;
    float vv[4] = {v.x, v.y, v.z, v.w};
#pragma unroll
    for (int c = 0; c < 4; ++c) {
      float f = vv[c];
      if (f > tv[TOPK - 1]) {
        tv[TOPK - 1] = f;
        ti[TOPK - 1] = base + c;
#pragma unroll
        for (int q = TOPK - 1; q > 0; --q) {
          if (tv[q] > tv[q - 1]) {  // strict > keeps earlier (lower) indices first on ties
            float fv = tv[q]; tv[q] = tv[q - 1]; tv[q - 1] = fv;
            int   fi = ti[q]; ti[q] = ti[q - 1]; ti[q - 1] = fi;
          } else break;
        }
      }
    }
  }

  __shared__ float sv[256 * TOPK];
  __shared__ int   si[256 * TOPK];
#pragma unroll
  for (int q = 0; q < TOPK; ++q) { sv[tid * TOPK + q] = tv[q]; si[tid * TOPK + q] = ti[q]; }
  __syncthreads();

  // Tree merge of sorted-descending 5-lists.
  for (int s = 128; s > 0; s >>= 1) {
    if (tid < s) {
      float* av = &sv[tid * TOPK];
      int*   ax = &si[tid * TOPK];
      float* bv = &sv[(tid + s) * TOPK];
      int*   bx = &si[(tid + s) * TOPK];
      float rv[TOPK]; int rx[TOPK];
      int ai = 0, bi = 0;
#pragma unroll
      for (int q = 0; q < TOPK; ++q) {
        float va = av[ai], vb = bv[bi];
        bool takeA = (va > vb) || (va == vb && ax[ai] <= bx[bi]);
        if (takeA) { rv[q] = va; rx[q] = ax[ai]; ++ai; }
        else       { rv[q] = vb; rx[q] = bx[bi]; ++bi; }
      }
#pragma unroll
      for (int q = 0; q < TOPK; ++q) { av[q] = rv[q]; ax[q] = rx[q]; }
    }
    __syncthreads();
  }

  if (tid < TOPK) idx_out[b * TOPK + tid] = si[tid];
}

// ---------------------------------------------------------------------------
// Kernel 2: gather + attention + weighted sum.  One block (128 thr = 4 waves)
// per row. Stage the 5 gathered sp_w rows + encoded row in LDS (b128 loads),
// contract the scores with chained V_WMMA_F32_16X16X4_F32 (exact f32, two
// independent accumulator chains per wave), softmax(5), weighted b128 sum.
//
// WMMA operand notes (all branchless — garbage in unread D rows/cols is OK
// since only D[m=0..4][n=0] (lane 0, VGPRs 0..4) is consumed):
//   A (16x4 f32, 2 VGPRs): lanes 0-15 {v0:K0, v1:K1}, lanes 16-31 {v0:K2,
//     v1:K3}; per-lane pair is CONTIGUOUS in d -> one ds_load_b64.
//   B (4x16 f32, 2 VGPRs): v0 {K0 lanes 0-15, K1 lanes 16-31}, v1 {K2, K3};
//     only n=0 (lanes 0 and 16) must be correct -> broadcast ds_load_b128 of
//     the enc chunk + two v_cndmask selects.
// ---------------------------------------------------------------------------
__global__ __launch_bounds__(128) void attn5_kernel(const float* __restrict__ sp_w,
                                                    const float* __restrict__ encoded,
                                                    const int*   __restrict__ idx_in,
                                                    float* __restrict__ out,
                                                    float* __restrict__ attn_out) {
  const int b    = blockIdx.x;
  const int tid  = threadIdx.x;
  const int lane = tid & 31;
  const int wave = tid >> 5;

  __shared__ __align__(16) float Wt[TOPK * DQ];  // gathered embeddings, 10 KB
  __shared__ __align__(16) float es[DQ];         // context encoding
  __shared__ float sdots[4][TOPK];               // per-wave partial dots
  __shared__ float sattn[TOPK];
  __shared__ int   sidx[TOPK];

  if (tid < TOPK) sidx[tid] = idx_in[b * TOPK + tid];
  __syncthreads();

  // Vectorized staging: b128 global loads -> b128 LDS stores.
  {
    const float4* spw4 = reinterpret_cast<const float4*>(sp_w);
    float4* Wt4 = reinterpret_cast<float4*>(Wt);
#pragma unroll
    for (int t = tid; t < TOPK * (DQ / 4); t += 128) {
      int k  = t >> 7;          // t / 128
      int d4 = t & 127;
      Wt4[k * 128 + d4] = spw4[(size_t)sidx[k] * (DQ / 4) + d4];
    }
    float4* es4 = reinterpret_cast<float4*>(es);
    es4[tid] = reinterpret_cast<const float4*>(encoded)[(size_t)b * (DQ / 4) + tid];
  }
  __syncthreads();

  // ---- scores[k] = sum_d Wt[k][d] * es[d] via chained f32 WMMA ----
  const int m     = lane & 15;
  const int hi    = lane >> 4;
  const int mm    = (m < TOPK) ? m : 0;       // clamp row index (rows >=5 unread)
  const bool l16  = (lane == 16);
  const float* arow = &Wt[mm * DQ + hi * 2];  // per-lane A base
  const float4* es4 = reinterpret_cast<const float4*>(es);

  v8f c0 = {}, c1 = {};
  const int db0 = wave * 128;                 // this wave's 128-wide D slice
#pragma unroll 4
  for (int ch = 0; ch < 32; ch += 2) {
    // chain 0
    {
      const int dbase = db0 + ch * 4;
      v2f a = *reinterpret_cast<const v2f*>(arow + dbase);  // ds_load_b64
      float4 e = es4[dbase >> 2];                           // broadcast ds_load_b128
      v2f bb;
      bb.x = l16 ? e.y : e.x;   // lane0: K0; lane16: K1; others: don't-care
      bb.y = l16 ? e.w : e.z;   // lane0: K2; lane16: K3; others: don't-care
      c0 = __builtin_amdgcn_wmma_f32_16x16x4_f32(false, a, false, bb, (short)0,
                                                 c0, false, false);
    }
    // chain 1 (independent accumulator -> overlaps WMMA latency)
    {
      const int dbase = db0 + (ch + 1) * 4;
      v2f a = *reinterpret_cast<const v2f*>(arow + dbase);
      float4 e = es4[dbase >> 2];
      v2f bb;
      bb.x = l16 ? e.y : e.x;
      bb.y = l16 ? e.w : e.z;
      c1 = __builtin_amdgcn_wmma_f32_16x16x4_f32(false, a, false, bb, (short)0,
                                                 c1, false, false);
    }
  }
  v8f c = c0 + c1;

  // D[m][n=0] lives at lane 0, VGPR m (m = 0..7 map to lanes 0-15).
  if (lane == 0) {
    sdots[wave][0] = c[0];
    sdots[wave][1] = c[1];
    sdots[wave][2] = c[2];
    sdots[wave][3] = c[3];
    sdots[wave][4] = c[4];
  }
  __syncthreads();

  if (tid == 0) {
    float dots[TOPK];
#pragma unroll
    for (int k = 0; k < TOPK; ++k)
      dots[k] = sdots[0][k] + sdots[1][k] + sdots[2][k] + sdots[3][k];
    float mx = dots[0];
#pragma unroll
    for (int k = 1; k < TOPK; ++k) mx = fmaxf(mx, dots[k]);
    float ex[TOPK], sum = 0.0f;
#pragma unroll
    for (int k = 0; k < TOPK; ++k) { ex[k] = __expf(dots[k] - mx); sum += ex[k]; }
    float inv = 1.0f / sum;
#pragma unroll
    for (int k = 0; k < TOPK; ++k) {
      float w = ex[k] * inv;
      sattn[k] = w;
      attn_out[b * TOPK + k] = w;
    }
  }
  __syncthreads();

  // out[b, :] = sum_k attn[k] * Wt[k, :]   (one float4 per thread, b128 store)
  {
    const float4* Wt4 = reinterpret_cast<const float4*>(Wt);
    float4 acc = make_float4(0.f, 0.f, 0.f, 0.f);
#pragma unroll
    for (int k = 0; k < TOPK; ++k) {
      float w = sattn[k];
      float4 x = Wt4[k * 128 + tid];
      acc.x += w * x.x; acc.y += w * x.y; acc.z += w * x.z; acc.w += w * x.w;
    }
    reinterpret_cast<float4*>(out)[(size_t)b * (DQ / 4) + tid] = acc;
  }
}

// ---------------------------------------------------------------------------
// d_out layout (tuple, flat, return order): out[B*D] f32 | attn[B*K] f32 |
// indices[B*K] int32 (bit-stored in the same 4-byte slots).
// ---------------------------------------------------------------------------
extern "C" void kernel_launch(void* const* d_in, const int* in_sizes, int n_in,
                              void* d_out, int out_size, void* d_ws, size_t ws_size,
                              hipStream_t stream) {
  (void)in_sizes; (void)n_in; (void)out_size; (void)d_ws; (void)ws_size;
  const float* sp_z = (const float*)d_in[0];
  const float* sp_w = (const float*)d_in[1];
  const float* enc  = (const float*)d_in[2];

  float* out      = (float*)d_out;                         // [B, D]
  float* attn_out = out + (size_t)BQ * DQ;                 // [B, K]
  int*   idx_out  = (int*)(attn_out + (size_t)BQ * TOPK);  // [B, K]

  topk5_kernel<<<BQ, 256, 0, stream>>>(sp_z, idx_out);
  attn5_kernel<<<BQ, 128, 0, stream>>>(sp_w, enc, idx_out, out, attn_out);
}